// RelativeTransformerModel_1400159338970
// MI455X (gfx1250) — compile-verified
//
#include <hip/hip_runtime.h>
#include <math.h>

typedef float v2f __attribute__((ext_vector_type(2)));
typedef float v8f __attribute__((ext_vector_type(8)));

#define NL 4
#define NB 4
#define NS 512
#define ND 512
#define NH 8
#define HD 64
#define D3 (3*ND)
#define LN_EPS 1e-5f

// ---------------- positional table: pos[l][d], l in [0,2S), d in [0,64) ----
__global__ __launch_bounds__(64) void pos_init_kernel(float* __restrict__ pos) {
  int l = blockIdx.x;      // 0..1023
  int d = threadIdx.x;     // 0..63
  int j = d & 31;
  float f = expf((float)j * (-9.210340371976184f / 31.0f));
  float p = (float)(l - NS);
  float a = p * f;
  pos[l * HD + d] = (d < 32) ? sinf(a) : cosf(a);
}

// ---------------- async global->LDS helpers (CDNA5 path) -------------------
__device__ __forceinline__ void async_g2l_b32(const float* gaddr, unsigned lds_off) {
  asm volatile("global_load_async_to_lds_b32 %0, %1, off"
               :: "v"(lds_off), "v"(gaddr) : "memory");
}
__device__ __forceinline__ void wait_async0() {
  asm volatile("s_wait_asynccnt 0" ::: "memory");
}

// ---------------- register-blocked fp32 WMMA GEMM --------------------------
// one wave computes an (IM*16) x (IN*16) tile: IMxIN accumulators,
// A/B loads amortized across the block -> ~16 FLOP/byte at IM=IN=4
template <int IM, int IN>
__global__ __launch_bounds__(32) void gemm_wmma_rb_kernel(
    const float* __restrict__ A, const float* __restrict__ Bm, float* __restrict__ C,
    int K, int lda, int ldb, int ldc, int tiles_n /* in units of IN*16 cols */,
    const float* __restrict__ bias, int act /*0=none,1=gelu(erf)*/) {
  int tile = blockIdx.x;
  int tn = (tile % tiles_n) * (IN * 16);
  int tm = (tile / tiles_n) * (IM * 16);
  int lane = threadIdx.x;
  int hi = lane >> 4, l15 = lane & 15, koff = hi << 1;

  const float* Arow[IM];
  const float* Bcol[IN];
#pragma unroll
  for (int im = 0; im < IM; ++im)
    Arow[im] = A + (size_t)(tm + im * 16 + l15) * lda;
#pragma unroll
  for (int in = 0; in < IN; ++in)
    Bcol[in] = Bm + (tn + in * 16 + l15);

  v8f acc[IM][IN] = {};
#pragma unroll 2
  for (int k0 = 0; k0 < K; k0 += 4) {
    v2f a[IM], b[IN];
#pragma unroll
    for (int im = 0; im < IM; ++im) {
      a[im].x = Arow[im][k0 + koff];
      a[im].y = Arow[im][k0 + koff + 1];
    }
#pragma unroll
    for (int in = 0; in < IN; ++in) {
      b[in].x = Bcol[in][(size_t)(k0 + koff) * ldb];
      b[in].y = Bcol[in][(size_t)(k0 + koff + 1) * ldb];
    }
#pragma unroll
    for (int im = 0; im < IM; ++im)
#pragma unroll
      for (int in = 0; in < IN; ++in)
        acc[im][in] = __builtin_amdgcn_wmma_f32_16x16x4_f32(
            false, a[im], false, b[in], (short)0, acc[im][in], false, false);
  }

#pragma unroll
  for (int in = 0; in < IN; ++in) {
    int n = tn + in * 16 + l15;
    float bv = bias ? bias[n] : 0.0f;
#pragma unroll
    for (int im = 0; im < IM; ++im) {
      int mb = tm + im * 16 + hi * 8;
#pragma unroll
      for (int r = 0; r < 8; ++r) {
        float v = acc[im][in][r] + bv;
        if (act == 1) v = 0.5f * v * (1.0f + erff(v * 0.70710678118654752f));
        C[(size_t)(mb + r) * ldc + n] = v;
      }
    }
  }
}

// ---------------- fused attention scores -----------------------------------
// attn[q,k] = ((q+rr)·k + (q+rw)·pos[S+k-q] + k·pos[S+q-k+2(k0-q0)]) / 8, masked; q0/k0 = tile bases.
// q/k/pos tiles staged into LDS with async global->LDS DMA (ASYNCcnt).
__global__ __launch_bounds__(32) void attn_scores_kernel(
    const float* __restrict__ qkv,            // [B][S][3D]
    const float* __restrict__ rr,             // [H][HD]
    const float* __restrict__ rw,             // [H][HD]
    const float* __restrict__ pos,            // [2S][HD]
    const unsigned char* __restrict__ mask,   // [B][S][S]
    float* __restrict__ scores) {             // [B*H][S][S]
  __shared__ float sQ[16][HD];
  __shared__ float sK[16][HD];
  __shared__ float sPos[31][HD];
  __shared__ float sR[2][HD];                 // [0]=rr, [1]=rw
  int bh = blockIdx.z, b = bh / NH, h = bh % NH;
  int q0 = blockIdx.y * 16, k0 = blockIdx.x * 16;
  int lane = threadIdx.x;
  const float* qbase = qkv + (size_t)b * NS * D3 + h * HD;
  const float* kbase = qbase + ND;

  unsigned ldsQ = (unsigned)(size_t)&sQ[0][0];
  unsigned ldsK = (unsigned)(size_t)&sK[0][0];
  unsigned ldsP = (unsigned)(size_t)&sPos[0][0];
  unsigned ldsR = (unsigned)(size_t)&sR[0][0];

  // q / k tiles: 16x64 each, 32 full-EXEC iterations of 1 dword/lane
  for (int i = lane; i < 16 * HD; i += 32) {
    int r = i >> 6, d = i & 63;
    async_g2l_b32(qbase + (size_t)(q0 + r) * D3 + d, ldsQ + 4u * i);
    async_g2l_b32(kbase + (size_t)(k0 + r) * D3 + d, ldsK + 4u * i);
  }
  // pos band of 31 contiguous rows (always inside [1,1023])
  int lbase = NS + k0 - q0 - 15;
  const float* pbase = pos + (size_t)lbase * HD;
  for (int i = lane; i < 31 * HD; i += 32)
    async_g2l_b32(pbase + i, ldsP + 4u * i);
  // rr / rw (64 floats each, contiguous per head)
  for (int i = lane; i < HD; i += 32) {
    async_g2l_b32(rr + h * HD + i, ldsR + 4u * i);
    async_g2l_b32(rw + h * HD + i, ldsR + 4u * (HD + i));
  }
  wait_async0();
  __syncthreads();

  int hi = lane >> 4, l15 = lane & 15, koff = hi << 1;
  v8f acc = {};
#pragma unroll
  for (int d0 = 0; d0 < HD; d0 += 4) {
    v2f a, bb;
    a.x = sQ[l15][d0 + koff] + sR[0][d0 + koff];
    a.y = sQ[l15][d0 + koff + 1] + sR[0][d0 + koff + 1];
    bb.x = sK[l15][d0 + koff];       // B[kk][n] = k[k0+n][kk] (transposed)
    bb.y = sK[l15][d0 + koff + 1];
    acc = __builtin_amdgcn_wmma_f32_16x16x4_f32(false, a, false, bb, (short)0, acc,
                                                false, false);
  }

  const float inv_scale = 0.125f;   // 1/sqrt(64)
#pragma unroll
  for (int r = 0; r < 8; ++r) {
    int m = r + hi * 8;             // local q
    int n = l15;                    // local k
    int q = q0 + m, k = k0 + n;
    int tB = 15 + (n - m);          // pos[S+k-q]
    int tE = 15 + (m - n);          // pos[S+k0-q0+m-n]
    float dotB = 0.f, dotE = 0.f;
    for (int d = 0; d < HD; ++d) {
      dotB = fmaf(sQ[m][d] + sR[1][d], sPos[tB][d], dotB);
      dotE = fmaf(sK[n][d], sPos[tE][d], dotE);
    }
    float val = (acc[r] + dotB + dotE) * inv_scale;
    if (!mask[((size_t)b * NS + q) * NS + k]) val = -1e9f;
    scores[((size_t)bh * NS + q) * NS + k] = val;
  }
}

// ---------------- row softmax with mask zeroing ----------------------------
__global__ __launch_bounds__(256) void softmax_kernel(
    float* __restrict__ sc, const unsigned char* __restrict__ mask) {
  __shared__ float wred[8];
  int row = blockIdx.x;             // B*H*S rows
  int q = row & (NS - 1);
  int bh = row >> 9;
  int b = bh / NH;
  float* sr = sc + (size_t)row * NS;
  const unsigned char* mr = mask + ((size_t)b * NS + q) * NS;
  int tid = threadIdx.x;
  float v0 = sr[tid], v1 = sr[tid + 256];
  float m = fmaxf(v0, v1);
  for (int o = 16; o; o >>= 1) m = fmaxf(m, __shfl_xor(m, o, 32));
  if ((tid & 31) == 0) wred[tid >> 5] = m;
  __syncthreads();
  float bm = wred[0];
  for (int i = 1; i < 8; ++i) bm = fmaxf(bm, wred[i]);
  __syncthreads();
  float e0 = expf(v0 - bm), e1 = expf(v1 - bm);
  float s = e0 + e1;
  for (int o = 16; o; o >>= 1) s += __shfl_xor(s, o, 32);
  if ((tid & 31) == 0) wred[tid >> 5] = s;
  __syncthreads();
  float ts = 0.f;
  for (int i = 0; i < 8; ++i) ts += wred[i];
  float inv = 1.0f / ts;
  sr[tid] = mr[tid] ? e0 * inv : 0.0f;
  sr[tid + 256] = mr[tid + 256] ? e1 * inv : 0.0f;
}

// ---------------- att @ v: 64x64 tile per wave, scattered into (B,S,D) -----
__global__ __launch_bounds__(32) void attn_v_kernel(
    const float* __restrict__ att,   // [B*H][S][S]
    const float* __restrict__ qkv,   // v at +2D + h*HD, row stride 3D
    float* __restrict__ o) {         // [B][S][D]
  int bh = blockIdx.y, b = bh / NH, h = bh % NH;
  int tm = blockIdx.x * 64;          // q tile base (S/64 = 8 tiles)
  int lane = threadIdx.x;
  int hi = lane >> 4, l15 = lane & 15, koff = hi << 1;
  const float* vbase = qkv + (size_t)b * NS * D3 + 2 * ND + h * HD;

  const float* Arow[4];
  const float* Bcol[4];
#pragma unroll
  for (int im = 0; im < 4; ++im)
    Arow[im] = att + ((size_t)bh * NS + tm + im * 16 + l15) * NS;
#pragma unroll
  for (int in = 0; in < 4; ++in)
    Bcol[in] = vbase + (in * 16 + l15);

  v8f acc[4][4] = {};
#pragma unroll 2
  for (int k0 = 0; k0 < NS; k0 += 4) {
    v2f a[4], bb[4];
#pragma unroll
    for (int im = 0; im < 4; ++im) {
      a[im].x = Arow[im][k0 + koff];
      a[im].y = Arow[im][k0 + koff + 1];
    }
#pragma unroll
    for (int in = 0; in < 4; ++in) {
      bb[in].x = Bcol[in][(size_t)(k0 + koff) * D3];
      bb[in].y = Bcol[in][(size_t)(k0 + koff + 1) * D3];
    }
#pragma unroll
    for (int im = 0; im < 4; ++im)
#pragma unroll
      for (int in = 0; in < 4; ++in)
        acc[im][in] = __builtin_amdgcn_wmma_f32_16x16x4_f32(
            false, a[im], false, bb[in], (short)0, acc[im][in], false, false);
  }
#pragma unroll
  for (int in = 0; in < 4; ++in) {
    int n = h * HD + in * 16 + l15;
#pragma unroll
    for (int im = 0; im < 4; ++im) {
      int mb = tm + im * 16 + hi * 8;
#pragma unroll
      for (int r = 0; r < 8; ++r)
        o[((size_t)b * NS + mb + r) * ND + n] = acc[im][in][r];
    }
  }
}

// ---------------- residual add + layernorm ---------------------------------
__global__ __launch_bounds__(256) void add_ln_kernel(
    const float* __restrict__ a, const float* __restrict__ b,
    const float* __restrict__ g, const float* __restrict__ be,
    float* __restrict__ out) {
  __shared__ float wred[8];
  int row = blockIdx.x;
  int tid = threadIdx.x;
  const float* ar = a + (size_t)row * ND;
  const float* br = b + (size_t)row * ND;
  float x0 = ar[tid] + br[tid];
  float x1 = ar[tid + 256] + br[tid + 256];
  float s = x0 + x1;
  for (int o = 16; o; o >>= 1) s += __shfl_xor(s, o, 32);
  if ((tid & 31) == 0) wred[tid >> 5] = s;
  __syncthreads();
  float tot = 0.f;
  for (int i = 0; i < 8; ++i) tot += wred[i];
  float mu = tot * (1.0f / ND);
  __syncthreads();
  float d0 = x0 - mu, d1 = x1 - mu;
  float vs = d0 * d0 + d1 * d1;
  for (int o = 16; o; o >>= 1) vs += __shfl_xor(vs, o, 32);
  if ((tid & 31) == 0) wred[tid >> 5] = vs;
  __syncthreads();
  float vtot = 0.f;
  for (int i = 0; i < 8; ++i) vtot += wred[i];
  float rstd = rsqrtf(vtot * (1.0f / ND) + LN_EPS);
  out[(size_t)row * ND + tid] = d0 * rstd * g[tid] + be[tid];
  out[(size_t)row * ND + tid + 256] = d1 * rstd * g[tid + 256] + be[tid + 256];
}

extern "C" void kernel_launch(void* const* d_in, const int* in_sizes, int n_in,
                              void* d_out, int out_size, void* d_ws, size_t ws_size,
                              hipStream_t stream) {
  (void)in_sizes; (void)n_in; (void)out_size; (void)ws_size;
  const float* x    = (const float*)d_in[0];
  const float* Wqkv = (const float*)d_in[1];
  const float* r_r  = (const float*)d_in[2];
  const float* r_w  = (const float*)d_in[3];
  const float* ln1g = (const float*)d_in[4];
  const float* ln1b = (const float*)d_in[5];
  const float* w1   = (const float*)d_in[6];
  const float* b1   = (const float*)d_in[7];
  const float* w2   = (const float*)d_in[8];
  const float* b2   = (const float*)d_in[9];
  const float* ln2g = (const float*)d_in[10];
  const float* ln2b = (const float*)d_in[11];
  const unsigned char* self_mask = (const unsigned char*)d_in[13];

  float* ws = (float*)d_ws;
  size_t off = 0;
  float* pos    = ws + off; off += (size_t)2 * NS * HD;
  float* qkvb   = ws + off; off += (size_t)NB * NS * D3;
  float* scores = ws + off; off += (size_t)NB * NH * NS * NS;
  float* obuf   = ws + off; off += (size_t)NB * NS * ND;
  float* h1     = ws + off; off += (size_t)NB * NS * ND;
  float* f1     = ws + off; off += (size_t)NB * NS * ND;
  float* f2     = ws + off; off += (size_t)NB * NS * ND;
  float* xbuf   = ws + off;                                   // total ~64 MB

  pos_init_kernel<<<2 * NS, 64, 0, stream>>>(pos);

  const int M = NB * NS;  // 2048
  for (int l = 0; l < NL; ++l) {
    const float* xin = (l == 0) ? x : xbuf;
    // QKV projection: (2048 x 512) @ (512 x 1536), 64x64 tiles -> 768 waves
    gemm_wmma_rb_kernel<4, 4><<<(M / 64) * (D3 / 64), 32, 0, stream>>>(
        xin, Wqkv + (size_t)l * ND * D3, qkvb, ND, ND, D3, D3, D3 / 64, nullptr, 0);
    // fused relative-position attention scores
    attn_scores_kernel<<<dim3(NS / 16, NS / 16, NB * NH), 32, 0, stream>>>(
        qkvb, r_r + (size_t)l * NH * HD, r_w + (size_t)l * NH * HD, pos, self_mask,
        scores);
    softmax_kernel<<<NB * NH * NS, 256, 0, stream>>>(scores, self_mask);
    attn_v_kernel<<<dim3(NS / 64, NB * NH), 32, 0, stream>>>(scores, qkvb, obuf);
    add_ln_kernel<<<M, 256, 0, stream>>>(obuf, xin, ln1g + (size_t)l * ND,
                                         ln1b + (size_t)l * ND, h1);
    // FFN1 + bias + exact GELU: (2048 x 512) @ (512 x 512)
    gemm_wmma_rb_kernel<4, 4><<<(M / 64) * (ND / 64), 32, 0, stream>>>(
        h1, w1 + (size_t)l * ND * ND, f1, ND, ND, ND, ND, ND / 64,
        b1 + (size_t)l * ND, 1);
    // FFN2 + bias
    gemm_wmma_rb_kernel<4, 4><<<(M / 64) * (ND / 64), 32, 0, stream>>>(
        f1, w2 + (size_t)l * ND * ND, f2, ND, ND, ND, ND, ND / 64,
        b2 + (size_t)l * ND, 0);
    add_ln_kernel<<<M, 256, 0, stream>>>(
        f2, h1, ln2g + (size_t)l * ND, ln2b + (size_t)l * ND,
        (l == NL - 1) ? (float*)d_out : xbuf);
  }
}